// GraphEncoder_28587302322919
// MI455X (gfx1250) — compile-verified
//
#include <hip/hip_runtime.h>
#include <hip/hip_bf16.h>

typedef __attribute__((ext_vector_type(2))) float v2f;
typedef __attribute__((ext_vector_type(8))) float v8f;

#define IN_DIM  384
#define HID_DIM 64

// ---------------------------------------------------------------------------
// init: agg[n][j] = bias[j]  (fuses the "+ b" of the reference into the
// scatter accumulator so no extra pass is needed afterwards)
// ---------------------------------------------------------------------------
__global__ void init_bias_kernel(float* __restrict__ out,
                                 const float* __restrict__ bias,
                                 long long total) {
    long long t = (long long)blockIdx.x * blockDim.x + threadIdx.x;
    if (t >= total) return;
    out[t] = bias[t & 63];
}

// ---------------------------------------------------------------------------
// GEMM1: h1 = (x * mask1 * 2) @ W1       [N,384] x [384,64] -> [N,64]
// One wave computes a 16x64 tile via 4 accumulators, K stepped by 4 using
// V_WMMA_F32_16X16X4_F32 (fp32 matrix pipe, reference precision).
// Dropout is fused branchlessly: b64 loads of x/mask, v_cndmask scale.
// ---------------------------------------------------------------------------
__global__ void gemm1_wmma_kernel(const float* __restrict__ x,
                                  const int*   __restrict__ mask1,
                                  const float* __restrict__ W1,
                                  float*       __restrict__ h1,
                                  int ntiles, int n_nodes) {
    const int wave = (int)(((long long)blockIdx.x * blockDim.x + threadIdx.x) >> 5);
    if (wave >= ntiles) return;
    const int lane  = threadIdx.x & 31;
    const int r0    = wave << 4;                 // first row of tile
    if (r0 + 16 > n_nodes) return;               // (N is a multiple of 16)
    const int m     = r0 + (lane & 15);          // A row for this lane
    const int khalf = (lane >> 4) << 1;          // 0 for lanes 0-15, 2 for 16-31
    const int col   = lane & 15;                 // B/D column within 16-wide frag

    v8f c0 = {}, c1 = {}, c2 = {}, c3 = {};
    const float* xrow = x     + (long long)m * IN_DIM + khalf;
    const int*   mrow_p = mask1 + (long long)m * IN_DIM + khalf;

    #pragma unroll 2
    for (int k = 0; k < IN_DIM; k += 4) {
        const int ka = k + khalf;
        // A fragment: unconditional b64 loads, branchless dropout scale
        const float2 xv = *(const float2*)(xrow + k);
        const int2   mv = *(const int2*)(mrow_p + k);
        v2f a;
        a.x = xv.x * (mv.x ? 2.0f : 0.0f);
        a.y = xv.y * (mv.y ? 2.0f : 0.0f);
        // B fragments: W1 rows ka, ka+1 (row-major [384,64]); L2-resident
        const float* Wk0 = W1 + (long long)(ka    ) * HID_DIM;
        const float* Wk1 = W1 + (long long)(ka + 1) * HID_DIM;
        v2f b0, b1, b2, b3;
        b0.x = Wk0[col     ]; b0.y = Wk1[col     ];
        b1.x = Wk0[col + 16]; b1.y = Wk1[col + 16];
        b2.x = Wk0[col + 32]; b2.y = Wk1[col + 32];
        b3.x = Wk0[col + 48]; b3.y = Wk1[col + 48];

        c0 = __builtin_amdgcn_wmma_f32_16x16x4_f32(false, a, false, b0, (short)0, c0, false, false);
        c1 = __builtin_amdgcn_wmma_f32_16x16x4_f32(false, a, false, b1, (short)0, c1, false, false);
        c2 = __builtin_amdgcn_wmma_f32_16x16x4_f32(false, a, false, b2, (short)0, c2, false, false);
        c3 = __builtin_amdgcn_wmma_f32_16x16x4_f32(false, a, false, b3, (short)0, c3, false, false);
    }

    // D layout: VGPR v holds row M = v (lanes 0-15) / v+8 (lanes 16-31)
    const int mrow = r0 + ((lane >> 4) << 3);
    #pragma unroll
    for (int v = 0; v < 8; ++v) {
        float* o = h1 + (long long)(mrow + v) * HID_DIM;
        o[col     ] = c0[v];
        o[col + 16] = c1[v];
        o[col + 32] = c2[v];
        o[col + 48] = c3[v];
    }
}

// ---------------------------------------------------------------------------
// GEMM2: h2 = (relu(agg1) * mask2 * 2) @ W2   [N,64] x [64,64] -> [N,64]
// agg1 already contains b1 (seeded by init_bias), so relu(agg1) == relu(agg+b1)
// ---------------------------------------------------------------------------
__global__ void gemm2_wmma_kernel(const float* __restrict__ agg1,
                                  const int*   __restrict__ mask2,
                                  const float* __restrict__ W2,
                                  float*       __restrict__ h2,
                                  int ntiles, int n_nodes) {
    const int wave = (int)(((long long)blockIdx.x * blockDim.x + threadIdx.x) >> 5);
    if (wave >= ntiles) return;
    const int lane  = threadIdx.x & 31;
    const int r0    = wave << 4;
    if (r0 + 16 > n_nodes) return;
    const int m     = r0 + (lane & 15);
    const int khalf = (lane >> 4) << 1;
    const int col   = lane & 15;

    v8f c0 = {}, c1 = {}, c2 = {}, c3 = {};
    const float* grow  = agg1  + (long long)m * HID_DIM + khalf;
    const int*   m2row = mask2 + (long long)m * HID_DIM + khalf;

    #pragma unroll 2
    for (int k = 0; k < HID_DIM; k += 4) {
        const int ka = k + khalf;
        const float2 gv = *(const float2*)(grow + k);
        const int2   mv = *(const int2*)(m2row + k);
        v2f a;
        a.x = fmaxf(gv.x, 0.0f) * (mv.x ? 2.0f : 0.0f);   // relu + dropout2, branchless
        a.y = fmaxf(gv.y, 0.0f) * (mv.y ? 2.0f : 0.0f);

        const float* Wk0 = W2 + (long long)(ka    ) * HID_DIM;
        const float* Wk1 = W2 + (long long)(ka + 1) * HID_DIM;
        v2f b0, b1, b2, b3;
        b0.x = Wk0[col     ]; b0.y = Wk1[col     ];
        b1.x = Wk0[col + 16]; b1.y = Wk1[col + 16];
        b2.x = Wk0[col + 32]; b2.y = Wk1[col + 32];
        b3.x = Wk0[col + 48]; b3.y = Wk1[col + 48];

        c0 = __builtin_amdgcn_wmma_f32_16x16x4_f32(false, a, false, b0, (short)0, c0, false, false);
        c1 = __builtin_amdgcn_wmma_f32_16x16x4_f32(false, a, false, b1, (short)0, c1, false, false);
        c2 = __builtin_amdgcn_wmma_f32_16x16x4_f32(false, a, false, b2, (short)0, c2, false, false);
        c3 = __builtin_amdgcn_wmma_f32_16x16x4_f32(false, a, false, b3, (short)0, c3, false, false);
    }

    const int mrow = r0 + ((lane >> 4) << 3);
    #pragma unroll
    for (int v = 0; v < 8; ++v) {
        float* o = h2 + (long long)(mrow + v) * HID_DIM;
        o[col     ] = c0[v];
        o[col + 16] = c1[v];
        o[col + 32] = c2[v];
        o[col + 48] = c3[v];
    }
}

// ---------------------------------------------------------------------------
// Edge scatter: agg[dst[e]] += h[src[e]] * w[e]
// 16 lanes per edge, float4 gather, hardware f32 atomics (accumulator is
// 25.6 MB -> fully L2-resident on the 192 MB L2).
// ---------------------------------------------------------------------------
__global__ void scatter_kernel(const float* __restrict__ h,
                               const int*   __restrict__ src,
                               const int*   __restrict__ dst,
                               const float* __restrict__ w,
                               float*       __restrict__ agg,
                               int n_edges) {
    long long t = (long long)blockIdx.x * blockDim.x + threadIdx.x;
    int e = (int)(t >> 4);
    if (e >= n_edges) return;
    int cq = (int)(t & 15) << 2;                 // 4-float column group
    int s = src[e], d = dst[e];
    float we = w[e];
    const float4 v = *(const float4*)(h + (long long)s * HID_DIM + cq);
    float* p = agg + (long long)d * HID_DIM + cq;
    atomicAdd(p + 0, v.x * we);
    atomicAdd(p + 1, v.y * we);
    atomicAdd(p + 2, v.z * we);
    atomicAdd(p + 3, v.w * we);
}

// ---------------------------------------------------------------------------
// launch
// ---------------------------------------------------------------------------
extern "C" void kernel_launch(void* const* d_in, const int* in_sizes, int n_in,
                              void* d_out, int out_size, void* d_ws, size_t ws_size,
                              hipStream_t stream) {
    const float* x     = (const float*)d_in[0];
    const int*   eidx  = (const int*)  d_in[1];   // [2, E]
    const float* ew    = (const float*)d_in[2];
    const float* W1    = (const float*)d_in[3];
    const float* b1    = (const float*)d_in[4];
    const float* W2    = (const float*)d_in[5];
    const float* b2    = (const float*)d_in[6];
    const int*   mask1 = (const int*)  d_in[7];
    const int*   mask2 = (const int*)  d_in[8];
    float*       out   = (float*)d_out;

    const int n_nodes = in_sizes[0] / IN_DIM;     // 100000
    const int n_edges = in_sizes[1] / 2;          // 1600000
    const int* src = eidx;
    const int* dst = eidx + n_edges;

    const long long feat = (long long)n_nodes * HID_DIM;   // 6.4M floats
    float* h1   = (float*)d_ws;
    float* agg1 = h1 + feat;
    float* h2   = agg1 + feat;

    const int ntiles = (n_nodes + 15) / 16;       // 6250
    const int gemm_threads = ntiles * 32;
    const int gemm_blocks  = (gemm_threads + 255) / 256;
    const int init_blocks  = (int)((feat + 255) / 256);
    const long long sc_threads = (long long)n_edges * 16;
    const int sc_blocks = (int)((sc_threads + 255) / 256);

    // layer 1: agg1 = b1; h1 = drop(x) @ W1; agg1 += scatter(h1)
    init_bias_kernel<<<init_blocks, 256, 0, stream>>>(agg1, b1, feat);
    gemm1_wmma_kernel<<<gemm_blocks, 256, 0, stream>>>(x, mask1, W1, h1, ntiles, n_nodes);
    scatter_kernel<<<sc_blocks, 256, 0, stream>>>(h1, src, dst, ew, agg1, n_edges);

    // layer 2: h2 = drop(relu(agg1)) @ W2; out = b2; out += scatter(h2)
    gemm2_wmma_kernel<<<gemm_blocks, 256, 0, stream>>>(agg1, mask2, W2, h2, ntiles, n_nodes);
    init_bias_kernel<<<init_blocks, 256, 0, stream>>>(out, b2, feat);
    scatter_kernel<<<sc_blocks, 256, 0, stream>>>(h2, src, dst, ew, out, n_edges);
}